// SparseDepthPrediction_42502996361505
// MI455X (gfx1250) — compile-verified
//
#include <hip/hip_runtime.h>
#include <math.h>

// Problem constants (compile-time from the reference)
#define Bn  32
#define Hn  256
#define Wn  1216
#define SHn 16
#define SWn 48
#define KH  30          // int(2*256/17)
#define KW  49          // int(2*1216/49)
#define OHn (Hn - KH + 1)   // 227
#define OWn (Wn - KW + 1)   // 1168
#define KR  242         // rows of Hcol actually used: hi[15]+KH = 212+30
#define KRP 244         // padded to multiple of 4 for WMMA K-steps

typedef __attribute__((ext_vector_type(2))) float v2f;
typedef __attribute__((ext_vector_type(8))) float v8f;

// ---------------------------------------------------------------------------
// Kernel 1: streaming zero-init of sparse output + channel-interleave
// transpose [B,2,H,W] -> [B,H,W,2].  Pure bandwidth, 128-bit accesses.
// ---------------------------------------------------------------------------
__global__ void zt_kernel(const float* __restrict__ dvf,
                          float* __restrict__ outSparse,
                          float* __restrict__ outDvfT) {
    const int NW4 = Wn / 4;
    int t = blockIdx.x * blockDim.x + threadIdx.x;
    if (t >= Bn * Hn * NW4) return;
    int w4 = t % NW4;
    int bh = t / NW4;          // b*Hn + h
    int b  = bh / Hn;
    int h  = bh % Hn;

    const float4* p = reinterpret_cast<const float4*>(dvf);
    float4 a = p[(((size_t)b * 2 + 0) * Hn + h) * NW4 + w4]; // channel 0 (x)
    float4 c = p[(((size_t)b * 2 + 1) * Hn + h) * NW4 + w4]; // channel 1 (y)

    // zero-init sparse depth image
    reinterpret_cast<float4*>(outSparse)[(size_t)bh * NW4 + w4] =
        make_float4(0.f, 0.f, 0.f, 0.f);

    // interleaved [B,H,W,2] output
    float4* ot = reinterpret_cast<float4*>(outDvfT);
    size_t base = ((size_t)bh * NW4 + w4) * 2;
    ot[base + 0] = make_float4(a.x, c.x, a.y, c.y);
    ot[base + 1] = make_float4(a.z, c.z, a.w, c.w);
}

// ---------------------------------------------------------------------------
// Kernel 2: per-batch sampled separable Gaussian conv (vertical pass via
// V_WMMA_F32_16X16X4_F32 with an LDS-materialized banded A-matrix),
// grid construction, bilinear grid-sample of gt, scatter into sparse image.
// One block per batch, fully LDS-resident (~113 KB of 320 KB).
// ---------------------------------------------------------------------------
__global__ void sample_kernel(const float* __restrict__ gt,
                              const float* __restrict__ dvf,
                              float* __restrict__ outSparse,
                              float* __restrict__ outGrid) {
    extern __shared__ float smem[];
    float* ghn  = smem;                  // 32 (uses KH=30)
    float* gwn  = smem + 32;             // 64 (uses KW=49)
    float* Alds = smem + 96;             // [KRP][16]  banded Gaussian A-matrix
    float* Hcol = Alds + KRP * 16;       // [2][KRP][SWn]
    float* Ssm  = Hcol + 2 * KRP * SWn;  // [2][SHn][SWn]

    const int b   = blockIdx.x;
    const int tid = threadIdx.x;
    const int T   = blockDim.x;

    // --- Gaussian taps (double precision, matches numpy float64 pipeline) --
    if (tid == 0) {
        double sH = (2.0 * Hn / (SHn + 1)) / 3.0;
        double sW = (2.0 * Wn / (SWn + 1)) / 3.0;
        double s = 0.0;
        for (int p = 0; p < KH; ++p) {
            double z = ((double)p - (KH - 1) * 0.5) / sH;
            double e = exp(-0.5 * z * z);
            ghn[p] = (float)e; s += e;
        }
        float inv = (float)(1.0 / s);
        for (int p = 0; p < KH; ++p) ghn[p] *= inv;
        s = 0.0;
        for (int q = 0; q < KW; ++q) {
            double z = ((double)q - (KW - 1) * 0.5) / sW;
            double e = exp(-0.5 * z * z);
            gwn[q] = (float)e; s += e;
        }
        inv = (float)(1.0 / s);
        for (int q = 0; q < KW; ++q) gwn[q] *= inv;
    }
    // zero the K-pad rows so WMMA padding contributes exactly 0
    for (int t = tid; t < 2 * (KRP - KR) * SWn; t += T) {
        int j  = t % SWn;
        int rr = (t / SWn) % (KRP - KR);
        int c  = t / (SWn * (KRP - KR));
        Hcol[((size_t)c * KRP + KR + rr) * SWn + j] = 0.0f;
    }
    __syncthreads();

    // --- Materialize banded A-matrix in fragment order: Alds[k][m] --------
    // A[m][k] = ghn[k - hi[m]] if in band else 0.  Constant across c / j-tile,
    // so the WMMA loop below needs only unpredicated ds_loads (no EXEC churn).
    for (int t = tid; t < KRP * 16; t += T) {
        int m = t & 15;
        int k = t >> 4;
        int d = k - (m * OHn) / SHn;
        Alds[t] = ((unsigned)d < (unsigned)KH) ? ghn[d] : 0.0f;
    }

    // --- Stage A: horizontal conv at the 48 sampled column windows --------
    for (int t = tid; t < 2 * KR * SWn; t += T) {
        int j = t % SWn;
        int r = (t / SWn) % KR;
        int c = t / (SWn * KR);
        int wj = (j * OWn) / SWn;                 // wi[j]
        const float* src = dvf + (((size_t)b * 2 + c) * Hn + r) * Wn + wj;
        float acc = 0.0f;
#pragma unroll 7
        for (int q = 0; q < KW; ++q) acc = fmaf(gwn[q], src[q], acc);
        Hcol[((size_t)c * KRP + r) * SWn + j] = acc;
    }
    __syncthreads();

    // --- Stage B: vertical conv as banded GEMM via f32 WMMA ---------------
    // S(16 x 48) = A(16 x KRP) x Hcol(KRP x 48), per channel; 6 waves,
    // one 16x16 output tile each (2 channels x 3 j-tiles).
    int wave = tid >> 5, lane = tid & 31;
    if (wave < 6) {
        int c  = wave / 3;
        int jt = wave % 3;
        int m     = lane & 15;                 // A row / B,D column within tile
        int khalf = (lane >> 4) << 1;          // 0 or 2 (K split across lane halves)
        const float* ap = Alds + m + khalf * 16;
        const float* bp = Hcol + (size_t)c * KRP * SWn + jt * 16 + m + khalf * SWn;
        v8f acc = {};
#pragma unroll 4
        for (int k0 = 0; k0 < KRP; k0 += 4) {
            v2f av, bv;
            av.x = ap[k0 * 16];
            av.y = ap[k0 * 16 + 16];
            bv.x = bp[k0 * SWn];
            bv.y = bp[k0 * SWn + SWn];
            acc = __builtin_amdgcn_wmma_f32_16x16x4_f32(
                false, av, false, bv, (short)0, acc, false, false);
        }
        int ibase = (lane >> 4) * 8;           // D: rows v (lanes 0-15) / v+8
#pragma unroll
        for (int v = 0; v < 8; ++v)
            Ssm[((size_t)c * SHn + (ibase + v)) * SWn + jt * 16 + m] = acc[v];
    }
    __syncthreads();

    // --- Stage C: grid, bilinear sample of gt, scatter --------------------
    for (int s = tid; s < SHn * SWn; s += T) {
        int i = s / SWn, j = s % SWn;
        float sx = Ssm[(0 * SHn + i) * SWn + j] + (float)(2 * j - (SWn - 1)) / (float)SWn;
        float sy = Ssm[(1 * SHn + i) * SWn + j] + (float)(2 * i - (SHn - 1)) / (float)SHn;
        size_t g = (((size_t)b * SHn + i) * SWn + j) * 2;
        outGrid[g + 0] = sx;
        outGrid[g + 1] = sy;

        // bilinear grid_sample, zeros padding, align_corners=False
        float x = (sx + 1.0f) * (Wn * 0.5f) - 0.5f;
        float y = (sy + 1.0f) * (Hn * 0.5f) - 0.5f;
        float x0 = floorf(x), y0 = floorf(y);
        float x1 = x0 + 1.0f, y1 = y0 + 1.0f;
        float wx1 = x - x0, wx0 = 1.0f - wx1;
        float wy1 = y - y0, wy0 = 1.0f - wy1;
        const float* gtb = gt + (size_t)b * Hn * Wn;
        auto samp = [&](float yc, float xc, float wgt) -> float {
            bool ok = (yc >= 0.0f) && (yc <= (float)(Hn - 1)) &&
                      (xc >= 0.0f) && (xc <= (float)(Wn - 1));
            int yi = (int)fminf(fmaxf(yc, 0.0f), (float)(Hn - 1));
            int xi = (int)fminf(fmaxf(xc, 0.0f), (float)(Wn - 1));
            return gtb[yi * Wn + xi] * (ok ? wgt : 0.0f);
        };
        float val = samp(y0, x0, wy0 * wx0) + samp(y0, x1, wy0 * wx1) +
                    samp(y1, x0, wy1 * wx0) + samp(y1, x1, wy1 * wx1);

        int row = (int)((sy + 1.0f) * 0.5f * (float)Hn);  // trunc toward zero
        int col = (int)((sx + 1.0f) * 0.5f * (float)Wn);
        row = min(max(row, 0), Hn - 1);
        col = min(max(col, 0), Wn - 1);
        if (row >= 96)  // rows 0..95 are zeroed in the reference
            outSparse[((size_t)b * Hn + row) * Wn + col] = val;
    }
}

// ---------------------------------------------------------------------------
extern "C" void kernel_launch(void* const* d_in, const int* in_sizes, int n_in,
                              void* d_out, int out_size, void* d_ws, size_t ws_size,
                              hipStream_t stream) {
    (void)in_sizes; (void)n_in; (void)out_size; (void)d_ws; (void)ws_size;
    const float* gt  = (const float*)d_in[0];   // [B,1,H,W]
    const float* dvf = (const float*)d_in[1];   // [B,2,H,W]

    float* out       = (float*)d_out;
    float* outSparse = out;                                   // B*H*W
    float* outDvfT   = out + (size_t)Bn * Hn * Wn;            // B*H*W*2
    float* outGrid   = outDvfT + (size_t)Bn * Hn * Wn * 2;    // B*SH*SW*2

    int nvec = Bn * Hn * (Wn / 4);
    zt_kernel<<<(nvec + 255) / 256, 256, 0, stream>>>(dvf, outSparse, outDvfT);

    size_t shb = (size_t)(32 + 64 + KRP * 16 + 2 * KRP * SWn + 2 * SHn * SWn)
                 * sizeof(float);
    sample_kernel<<<Bn, 512, shb, stream>>>(gt, dvf, outSparse, outGrid);
}